// GraphEncoder_4621384810820
// MI455X (gfx1250) — compile-verified
//
#include <hip/hip_runtime.h>

typedef __attribute__((ext_vector_type(2))) float v2f;
typedef __attribute__((ext_vector_type(8))) float v8f;

// ---------------------------------------------------------------------------
// Utility kernels
// ---------------------------------------------------------------------------
__global__ void fill_kernel(float* __restrict__ p, float v, long n) {
  long i = (long)blockIdx.x * blockDim.x + threadIdx.x;
  if (i < n) p[i] = v;
}

__global__ void deg_accum_kernel(const int* __restrict__ dst, float* __restrict__ deg, int E) {
  int e = blockIdx.x * blockDim.x + threadIdx.x;
  if (e < E) atomicAdd(&deg[dst[e]], 1.0f);
}

__global__ void rsqrt_kernel(float* __restrict__ d, int n) {
  int i = blockIdx.x * blockDim.x + threadIdx.x;
  if (i < n) d[i] = rsqrtf(d[i]);
}

// ---------------------------------------------------------------------------
// f32 WMMA GEMM: C[nrows x NOUT] = A[nrows x 128] * W[128 x NOUT]
// One block = 16 rows, NOUT/16 waves, each wave owns a 16x16 output tile.
// A tile staged in LDS with padded stride (132) -> conflict-free WMMA feeds.
// A-operand lane map (ISA 7.12.2, 32-bit A 16x4): m = lane&15,
//   k = 2*(lane>>4) + vgpr.  B: n = lane&15, k = 2*(lane>>4) + vgpr.
// D: vgpr r -> m = r + 8*(lane>>4), n = lane&15.
// ---------------------------------------------------------------------------
template <int NOUT>
__global__ __launch_bounds__(32 * (NOUT / 16))
void gemm_wmma_kernel(const float* __restrict__ A, const float* __restrict__ W,
                      float* __restrict__ C, int nrows) {
  constexpr int K = 128;
  constexpr int LDSK = K + 4;                 // pad: stride 132 -> banks 4m+k, no conflicts
  constexpr int NTHREADS = 32 * (NOUT / 16);
  __shared__ float sA[16 * LDSK];

  const int row0 = blockIdx.x * 16;
  const int tid  = threadIdx.x;
  const int lane = tid & 31;
  const int w    = tid >> 5;

  // Cooperative float4 load of the 16 x 128 A tile (coalesced)
  for (int i = tid; i < 16 * (K / 4); i += NTHREADS) {
    const int r  = i >> 5;   // 32 float4 per row
    const int c4 = i & 31;
    float4 v = make_float4(0.f, 0.f, 0.f, 0.f);
    if (row0 + r < nrows)
      v = ((const float4*)(A + (size_t)(row0 + r) * K))[c4];
    float* dp = &sA[r * LDSK + c4 * 4];
    dp[0] = v.x; dp[1] = v.y; dp[2] = v.z; dp[3] = v.w;
  }
  __syncthreads();

  const int m     = lane & 15;
  const int khalf = (lane >> 4) * 2;
  const int n0    = w * 16;
  const int n     = n0 + m;

  v8f acc = {};
#pragma unroll
  for (int kk = 0; kk < K; kk += 4) {
    const int k = kk + khalf;
    v2f a, b;
    a.x = sA[m * LDSK + k];
    a.y = sA[m * LDSK + k + 1];
    b.x = W[(size_t)k * NOUT + n];
    b.y = W[(size_t)(k + 1) * NOUT + n];
    acc = __builtin_amdgcn_wmma_f32_16x16x4_f32(false, a, false, b, (short)0, acc,
                                                false, false);
  }

  const int mbase = (lane >> 4) * 8;
#pragma unroll
  for (int r = 0; r < 8; ++r) {
    const int mm = mbase + r;
    if (row0 + mm < nrows)
      C[(size_t)(row0 + mm) * NOUT + n] = acc[r];
  }
}

// ---------------------------------------------------------------------------
// Edge aggregation, F=128: one wave per edge; lane loads float4 of h[src],
// scales by dinv[src]*dinv[dst], atomic-adds into agg[dst]. agg (51 MB) is
// L2-resident on MI455X (192 MB L2) so the atomics resolve in L2.
// ---------------------------------------------------------------------------
__global__ void edge_agg128_kernel(const float* __restrict__ h, const float* __restrict__ dinv,
                                   const int* __restrict__ src, const int* __restrict__ dst,
                                   float* __restrict__ agg, int E) {
  const int  lane = threadIdx.x & 31;
  const long e    = ((long)blockIdx.x * blockDim.x + threadIdx.x) >> 5;
  if (e >= E) return;
  const int   s    = src[e];
  const int   d    = dst[e];
  const float norm = dinv[s] * dinv[d];
  const float4 v   = ((const float4*)(h + (size_t)s * 128))[lane];
  float* a = agg + (size_t)d * 128 + lane * 4;
  atomicAdd(a + 0, v.x * norm);
  atomicAdd(a + 1, v.y * norm);
  atomicAdd(a + 2, v.z * norm);
  atomicAdd(a + 3, v.w * norm);
}

// F=64: two edges per wave (16 lanes each) to keep all lanes busy.
__global__ void edge_agg64_kernel(const float* __restrict__ h, const float* __restrict__ dinv,
                                  const int* __restrict__ src, const int* __restrict__ dst,
                                  float* __restrict__ agg, int E) {
  const int  lane = threadIdx.x & 31;
  const int  sub  = lane & 15;
  const long wid  = ((long)blockIdx.x * blockDim.x + threadIdx.x) >> 5;
  const long e    = wid * 2 + (lane >> 4);
  if (e >= E) return;
  const int   s    = src[e];
  const int   d    = dst[e];
  const float norm = dinv[s] * dinv[d];
  const float4 v   = ((const float4*)(h + (size_t)s * 64))[sub];
  float* a = agg + (size_t)d * 64 + sub * 4;
  atomicAdd(a + 0, v.x * norm);
  atomicAdd(a + 1, v.y * norm);
  atomicAdd(a + 2, v.z * norm);
  atomicAdd(a + 3, v.w * norm);
}

// out1 = relu(agg1 + h1*dinv^2 + b1), in place into agg1 (F = 128)
__global__ void combine_relu_kernel(float* __restrict__ agg, const float* __restrict__ h,
                                    const float* __restrict__ dinv,
                                    const float* __restrict__ bias, long total) {
  long idx = (long)blockIdx.x * blockDim.x + threadIdx.x;
  if (idx >= total) return;
  const int   i  = (int)(idx >> 7);
  const int   f  = (int)(idx & 127);
  const float di = dinv[i];
  const float v  = agg[idx] + h[idx] * di * di + bias[f];
  agg[idx] = v > 0.f ? v : 0.f;
}

// out = out(=agg2) + h2*dinv^2 + b2 (F = 64, no relu)
__global__ void final_combine_kernel(float* __restrict__ out, const float* __restrict__ h2,
                                     const float* __restrict__ dinv,
                                     const float* __restrict__ bias, long total) {
  long idx = (long)blockIdx.x * blockDim.x + threadIdx.x;
  if (idx >= total) return;
  const int   i  = (int)(idx >> 6);
  const int   f  = (int)(idx & 63);
  const float di = dinv[i];
  out[idx] = out[idx] + h2[idx] * di * di + bias[f];
}

// ---------------------------------------------------------------------------
static inline unsigned cdiv(long a, long b) { return (unsigned)((a + b - 1) / b); }

extern "C" void kernel_launch(void* const* d_in, const int* in_sizes, int n_in,
                              void* d_out, int out_size, void* d_ws, size_t ws_size,
                              hipStream_t stream) {
  const float* x  = (const float*)d_in[0];
  const int*   ei = (const int*)d_in[1];
  const float* W1 = (const float*)d_in[2];
  const float* b1 = (const float*)d_in[3];
  const float* W2 = (const float*)d_in[4];
  const float* b2 = (const float*)d_in[5];

  const int IN = 128, HID = 128, OUT = 64;
  const int N = in_sizes[0] / IN;
  const int E = in_sizes[1] / 2;
  const int* srcI = ei;       // edge_index[0]
  const int* dstI = ei + E;   // edge_index[1]
  float* out = (float*)d_out;

  // Workspace layout (floats): dinv[N] | h1[N*128] | agg1[N*128]; h2 reuses h1.
  float* dinv = (float*)d_ws;
  float* h1   = dinv + N;
  float* agg1 = h1 + (size_t)N * HID;
  float* h2   = h1;  // safe: h1 dead after combine_relu, before gemm2 writes

  const int BLK = 256;
  const long nh1 = (long)N * HID;
  const long nh2 = (long)N * OUT;

  // 1) degree (self-loop = 1.0), then dinv = rsqrt(deg)
  fill_kernel<<<cdiv(N, BLK), BLK, 0, stream>>>(dinv, 1.0f, N);
  deg_accum_kernel<<<cdiv(E, BLK), BLK, 0, stream>>>(dstI, dinv, E);
  rsqrt_kernel<<<cdiv(N, BLK), BLK, 0, stream>>>(dinv, N);

  // 2) zero accumulators (must happen every call: harness replays the graph)
  fill_kernel<<<cdiv(nh1, BLK), BLK, 0, stream>>>(agg1, 0.0f, nh1);
  fill_kernel<<<cdiv(nh2, BLK), BLK, 0, stream>>>(out, 0.0f, nh2);

  // 3) h1 = X @ W1  (WMMA f32)
  gemm_wmma_kernel<128><<<cdiv(N, 16), 256, 0, stream>>>(x, W1, h1, N);

  // 4) agg1 += norm * h1[src]
  edge_agg128_kernel<<<cdiv((long)E * 32, BLK), BLK, 0, stream>>>(h1, dinv, srcI, dstI, agg1, E);

  // 5) out1 = relu(agg1 + h1*dinv^2 + b1)   (in place -> agg1)
  combine_relu_kernel<<<cdiv(nh1, BLK), BLK, 0, stream>>>(agg1, h1, dinv, b1, nh1);

  // 6) h2 = out1 @ W2  (WMMA f32)
  gemm_wmma_kernel<64><<<cdiv(N, 16), 128, 0, stream>>>(agg1, W2, h2, N);

  // 7) out += norm * h2[src]
  edge_agg64_kernel<<<cdiv(((long)E + 1) / 2 * 32, BLK), BLK, 0, stream>>>(h2, dinv, srcI, dstI, out, E);

  // 8) out += h2*dinv^2 + b2
  final_combine_kernel<<<cdiv(nh2, BLK), BLK, 0, stream>>>(out, h2, dinv, b2, nh2);
}